// QMNAblation_54838142435823
// MI455X (gfx1250) — compile-verified
//
#include <hip/hip_runtime.h>
#include <math.h>

#define B_ 32
#define T_ 128
#define D_ 64
#define LDV 65   // LDS row stride (odd -> conflict-free row & column access)

typedef __attribute__((ext_vector_type(2))) float v2f;
typedef __attribute__((ext_vector_type(8))) float v8f;

__device__ inline v8f wmma4(v2f a, v2f b, v8f c) {
  // V_WMMA_F32_16X16X4_F32 : D = A(16x4) * B(4x16) + C(16x16), fp32 throughout
  return __builtin_amdgcn_wmma_f32_16x16x4_f32(false, a, false, b, (short)0, c,
                                               false, false);
}

// Complex 16x16 output tile of Y = A * B, A/B complex 64x64 in LDS (row stride LDV).
// BMODE: 0 = B as stored (row-major KxN), 1 = B = conj(U)^T from U, 2 = B = U^T (no conj)
template <int BMODE>
__device__ inline void cmm_tile(const float* __restrict__ Ar, const float* __restrict__ Ai,
                                const float* __restrict__ Br, const float* __restrict__ Bi,
                                int i0, int j0, int r15, int hl, v8f& accR, v8f& accI) {
#pragma unroll
  for (int k = 0; k < 64; k += 4) {
    // A tile: rows i0..i0+15, K = k..k+3.  lane: row = i0+r15, ks = k + 2*hl (+1 in .y)
    int aoff = (i0 + r15) * LDV + k + 2 * hl;
    v2f a_r; a_r.x = Ar[aoff]; a_r.y = Ar[aoff + 1];
    v2f a_i; a_i.x = Ai[aoff]; a_i.y = Ai[aoff + 1];
    v2f b_r, b_i;
    if (BMODE == 0) {
      int b0 = (k + 2 * hl) * LDV + j0 + r15;
      b_r.x = Br[b0]; b_r.y = Br[b0 + LDV];
      b_i.x = Bi[b0]; b_i.y = Bi[b0 + LDV];
    } else {
      int boff = (j0 + r15) * LDV + k + 2 * hl;   // transposed read of U
      b_r.x = Br[boff]; b_r.y = Br[boff + 1];
      float s = (BMODE == 1) ? -1.0f : 1.0f;      // conj for U^H
      b_i.x = s * Bi[boff]; b_i.y = s * Bi[boff + 1];
    }
    v2f a_in; a_in.x = -a_i.x; a_in.y = -a_i.y;   // f32 WMMA NEG bits are C-only
    accR = wmma4(a_r, b_r, accR);   // Yr += Ar*Br
    accR = wmma4(a_in, b_i, accR);  // Yr -= Ai*Bi
    accI = wmma4(a_r, b_i, accI);   // Yi += Ar*Bi
    accI = wmma4(a_i, b_r, accI);   // Yi += Ai*Br
  }
}

// ---------------------------------------------------------------------------
// Kernel 1: per-(b,t) prep. reps -> softmax weights, speaker phase, psi,
// v_m = Ux @ psi_m  (rank-3 factorization of Ux rho Ux^H).
// ---------------------------------------------------------------------------
__global__ __launch_bounds__(64) void prep_kernel(
    const float* __restrict__ mod0, const float* __restrict__ Wp0, const float* __restrict__ bp0,
    const float* __restrict__ mod1, const float* __restrict__ Wp1, const float* __restrict__ bp1,
    const float* __restrict__ mod2, const float* __restrict__ Wp2, const float* __restrict__ bp2,
    const float* __restrict__ smask, const float* __restrict__ phase_table,
    const float* __restrict__ Ux_r, const float* __restrict__ Ux_i,
    float* __restrict__ Vr, float* __restrict__ Vi, float* __restrict__ Wt) {
  __shared__ float sx[304];
  __shared__ float sRep[3][64];
  __shared__ float sPsiR[64], sPsiI[64];
  __shared__ float sNorm[3];

  const int bt = blockIdx.x;
  const int t = bt & (T_ - 1);
  const int j = threadIdx.x;

  const float* mods[3] = {mod0, mod1, mod2};
  const float* Wps[3] = {Wp0, Wp1, Wp2};
  const float* bps[3] = {bp0, bp1, bp2};
  const int dms[3] = {300, 74, 35};

  for (int m = 0; m < 3; ++m) {
    const int dm = dms[m];
    for (int idx = j; idx < dm; idx += 64) sx[idx] = mods[m][(long)bt * dm + idx];
    __syncthreads();
    float acc = bps[m][j];
    for (int k = 0; k < dm; ++k) acc += sx[k] * Wps[m][k * 64 + j];
    sRep[m][j] = fmaxf(acc, 0.0f);
    __syncthreads();
  }
  if (j < 3) {
    float s = 0.0f;
    for (int q = 0; q < 64; ++q) { float r = sRep[j][q]; s += r * r; }
    sNorm[j] = sqrtf(s);
  }
  __syncthreads();

  // softmax over modality norms (redundant per-thread, 3 elements)
  float n0 = sNorm[0], n1 = sNorm[1], n2 = sNorm[2];
  float mx = fmaxf(n0, fmaxf(n1, n2));
  float e0 = expf(n0 - mx), e1 = expf(n1 - mx), e2 = expf(n2 - mx);
  float es = e0 + e1 + e2;
  float wgt[3] = {e0 / es, e1 / es, e2 / es};

  // speaker-conditioned phase (argmax ties -> first, matches jnp.argmax)
  float s0 = smask[bt * 2 + 0], s1 = smask[bt * 2 + 1];
  int sid = (s1 > s0) ? 1 : 0;
  float freq = powf(10000.0f, -(float)j * (1.0f / 64.0f));
  float phase = (float)t * freq + phase_table[sid * 64 + j] * 6.28f;  // repo uses 2*3.14
  float cp = cosf(phase), sp = sinf(phase);

  for (int m = 0; m < 3; ++m) {
    float nrm = fmaxf(sNorm[m], 1e-12f);
    float amp = sRep[m][j] / nrm;
    sPsiR[j] = amp * cp;
    sPsiI[j] = amp * sp;
    __syncthreads();
    float vr = 0.0f, vi = 0.0f;
    for (int k = 0; k < 64; ++k) {
      float ur = Ux_r[j * 64 + k], ui = Ux_i[j * 64 + k];
      float pr = sPsiR[k], pi = sPsiI[k];
      vr += ur * pr - ui * pi;
      vi += ur * pi + ui * pr;
    }
    long o = ((long)bt * 3 + m) * 64 + j;
    Vr[o] = vr;
    Vi[o] = vi;
    __syncthreads();
  }
  if (j < 3) Wt[bt * 3 + j] = wgt[j];
}

// ---------------------------------------------------------------------------
// Kernel 2: per-batch sequential scan. One block per b (32 blocks, 8 wave32s).
// Per step: M = Uh*h ; h = lam*rank3(v) + (1-lam)*M*Uh^H ; probs via h*K^T.
// All 64x64 complex matmuls run on V_WMMA_F32_16X16X4_F32.
// ---------------------------------------------------------------------------
__global__ __launch_bounds__(256) void scan_kernel(
    const float* __restrict__ Vr, const float* __restrict__ Vi, const float* __restrict__ Wt,
    const float* __restrict__ Uh_r, const float* __restrict__ Uh_i,
    const float* __restrict__ K_r, const float* __restrict__ K_i,
    const float* __restrict__ Lambda, float* __restrict__ probs) {
  __shared__ float sUhr[64 * LDV], sUhi[64 * LDV];
  __shared__ float sKr[64 * LDV], sKi[64 * LDV];
  __shared__ float shr[64 * LDV], shi[64 * LDV];
  __shared__ float sMr[64 * LDV], sMi[64 * LDV];
  __shared__ float sVr[3 * 64], sVi[3 * 64], sW[3];
  __shared__ float sProb[64], sCA[64];

  const int tid = threadIdx.x;
  const int lane = tid & 31, wave = tid >> 5;
  const int r15 = lane & 15, hl = lane >> 4;
  const int b = blockIdx.x;
  const float lam = Lambda[0];
  const float BETA = 0.8f;

  for (int idx = tid; idx < 64 * 64; idx += 256) {
    int r = idx >> 6, c = idx & 63;
    sUhr[r * LDV + c] = Uh_r[idx];
    sUhi[r * LDV + c] = Uh_i[idx];
    sKr[r * LDV + c] = K_r[idx];
    sKi[r * LDV + c] = K_i[idx];
    shr[r * LDV + c] = (r == c) ? (1.0f / 64.0f) : 0.0f;  // h0 = I/d
    shi[r * LDV + c] = 0.0f;
  }
  __syncthreads();
  if (tid < 64) {  // |a_j|^2 / d   (measurement of the (1-beta)*I/d term)
    float s = 0.0f;
    for (int m = 0; m < 64; ++m) {
      float kr = sKr[tid * LDV + m], ki = sKi[tid * LDV + m];
      s += kr * kr + ki * ki;
    }
    sCA[tid] = s * (1.0f / 64.0f);
  }

  for (int t = 0; t < T_; ++t) {
    const long base = ((long)(b * T_ + t)) * 3 * 64;
    for (int idx = tid; idx < 192; idx += 256) {
      sVr[idx] = Vr[base + idx];
      sVi[idx] = Vi[base + idx];
    }
    if (tid < 3) sW[tid] = Wt[(b * T_ + t) * 3 + tid];
    if (tid < 64) sProb[tid] = 0.0f;
    __syncthreads();

    // ---- M = Uh @ h ----
#pragma unroll
    for (int tt = 0; tt < 2; ++tt) {
      int tile = 2 * wave + tt;
      int i0 = (tile >> 2) << 4, j0 = (tile & 3) << 4;
      v8f accR = {}, accI = {};
      cmm_tile<0>(sUhr, sUhi, shr, shi, i0, j0, r15, hl, accR, accI);
#pragma unroll
      for (int g = 0; g < 8; ++g) {
        int row = i0 + g + 8 * hl, col = j0 + r15;
        sMr[row * LDV + col] = accR[g];
        sMi[row * LDV + col] = accI[g];
      }
    }
    __syncthreads();

    // ---- h_new = lam * (sum_m w_m v v^H) + (1-lam) * M @ Uh^H ----
#pragma unroll
    for (int tt = 0; tt < 2; ++tt) {
      int tile = 2 * wave + tt;
      int i0 = (tile >> 2) << 4, j0 = (tile & 3) << 4;
      v8f accR = {}, accI = {};
      cmm_tile<1>(sMr, sMi, sUhr, sUhi, i0, j0, r15, hl, accR, accI);
#pragma unroll
      for (int g = 0; g < 8; ++g) {
        int row = i0 + g + 8 * hl, col = j0 + r15;
        float hxr = 0.0f, hxi = 0.0f;
#pragma unroll
        for (int m = 0; m < 3; ++m) {
          float ar = sVr[m * 64 + row], ai = sVi[m * 64 + row];
          float br = sVr[m * 64 + col], bi = sVi[m * 64 + col];
          float w = sW[m];
          hxr += w * (ar * br + ai * bi);   // Re(v_i * conj(v_j))
          hxi += w * (ai * br - ar * bi);   // Im(v_i * conj(v_j))
        }
        shr[row * LDV + col] = lam * hxr + (1.0f - lam) * accR[g];
        shi[row * LDV + col] = lam * hxi + (1.0f - lam) * accI[g];
      }
    }
    __syncthreads();

    // ---- measurement: Z = h_new @ K^T ; p_j = sum_i Re(conj(K[j,i]) Z[i,j]) ----
#pragma unroll
    for (int tt = 0; tt < 2; ++tt) {
      int tile = 2 * wave + tt;
      int i0 = (tile >> 2) << 4, j0 = (tile & 3) << 4;
      v8f accR = {}, accI = {};
      cmm_tile<2>(shr, shi, sKr, sKi, i0, j0, r15, hl, accR, accI);
      float p = 0.0f;
      int col = j0 + r15;
#pragma unroll
      for (int g = 0; g < 8; ++g) {
        int row = i0 + g + 8 * hl;
        p += sKr[col * LDV + row] * accR[g] + sKi[col * LDV + row] * accI[g];
      }
      atomicAdd(&sProb[col], p);
    }
    __syncthreads();

    if (tid < 64)
      probs[((b * T_ + t) << 6) + tid] = BETA * sProb[tid] + (1.0f - BETA) * sCA[tid];
  }
}

// ---------------------------------------------------------------------------
// Kernel 3: SimpleNet head + log_softmax over 6 classes.
// ---------------------------------------------------------------------------
__global__ __launch_bounds__(64) void head_kernel(
    const float* __restrict__ probs, const float* __restrict__ W1, const float* __restrict__ b1,
    const float* __restrict__ W2, const float* __restrict__ b2, float* __restrict__ out) {
  __shared__ float sp[64];
  __shared__ float sh1[64];
  __shared__ float slg[6];
  const int bt = blockIdx.x;
  const int j = threadIdx.x;
  sp[j] = probs[(bt << 6) + j];
  __syncthreads();
  float acc = b1[j];
  for (int k = 0; k < 64; ++k) acc += sp[k] * W1[k * 64 + j];
  sh1[j] = fmaxf(acc, 0.0f);
  __syncthreads();
  if (j < 6) {
    float a = b2[j];
    for (int k = 0; k < 64; ++k) a += sh1[k] * W2[k * 6 + j];
    slg[j] = tanhf(a);
  }
  __syncthreads();
  if (j < 6) {
    float mx = slg[0];
    for (int c = 1; c < 6; ++c) mx = fmaxf(mx, slg[c]);
    float se = 0.0f;
    for (int c = 0; c < 6; ++c) se += expf(slg[c] - mx);
    out[bt * 6 + j] = slg[j] - mx - logf(se);
  }
}

extern "C" void kernel_launch(void* const* d_in, const int* in_sizes, int n_in,
                              void* d_out, int out_size, void* d_ws, size_t ws_size,
                              hipStream_t stream) {
  const float* mod0 = (const float*)d_in[0];
  const float* Wp0 = (const float*)d_in[1];
  const float* bp0 = (const float*)d_in[2];
  const float* mod1 = (const float*)d_in[3];
  const float* Wp1 = (const float*)d_in[4];
  const float* bp1 = (const float*)d_in[5];
  const float* mod2 = (const float*)d_in[6];
  const float* Wp2 = (const float*)d_in[7];
  const float* bp2 = (const float*)d_in[8];
  const float* smask = (const float*)d_in[9];
  // d_in[10] = umask (unused by the reference)
  const float* phase_table = (const float*)d_in[11];
  const float* Uh_r = (const float*)d_in[12];
  const float* Uh_i = (const float*)d_in[13];
  const float* Ux_r = (const float*)d_in[14];
  const float* Ux_i = (const float*)d_in[15];
  const float* Lambda = (const float*)d_in[16];
  const float* K_r = (const float*)d_in[17];
  const float* K_i = (const float*)d_in[18];
  const float* W1 = (const float*)d_in[19];
  const float* b1 = (const float*)d_in[20];
  const float* W2 = (const float*)d_in[21];
  const float* b2 = (const float*)d_in[22];

  float* ws = (float*)d_ws;
  float* Vr = ws;                       // B*T*3*64
  float* Vi = Vr + B_ * T_ * 3 * 64;    // B*T*3*64
  float* Wt = Vi + B_ * T_ * 3 * 64;    // B*T*3
  float* pr = Wt + B_ * T_ * 3;         // B*T*64

  prep_kernel<<<B_ * T_, 64, 0, stream>>>(mod0, Wp0, bp0, mod1, Wp1, bp1, mod2, Wp2, bp2,
                                          smask, phase_table, Ux_r, Ux_i, Vr, Vi, Wt);
  scan_kernel<<<B_, 256, 0, stream>>>(Vr, Vi, Wt, Uh_r, Uh_i, K_r, K_i, Lambda, pr);
  head_kernel<<<B_ * T_, 64, 0, stream>>>(pr, W1, b1, W2, b2, (float*)d_out);
}